// UOTHead_79087527788856
// MI455X (gfx1250) — compile-verified
//
#include <hip/hip_runtime.h>
#include <math.h>

#define BATCH 8
#define NPTS 2048
#define MPTS 2048
#define CDIM 256
#define UOT_ITERS 5
#define LPAD 20            // floats per LDS row: 80B = 16B-aligned, conflict-free

typedef float v2f __attribute__((ext_vector_type(2)));
typedef float v8f __attribute__((ext_vector_type(8)));

// ---------------------------------------------------------------- init ------
__global__ void uot_init(const float* __restrict__ log_eps,
                         const float* __restrict__ log_gamma,
                         float* __restrict__ avec, float* __restrict__ scal) {
  int idx = blockIdx.x * blockDim.x + threadIdx.x;
  if (idx < BATCH * NPTS) avec[idx] = 1.0f / (float)NPTS;
  if (idx == 0) {
    float eps = expf(log_eps[0]) + 0.03f;
    float g   = expf(log_gamma[0]);
    scal[0] = eps;
    scal[1] = g / (g + eps);   // power
  }
}

// ------------------------------------------------------------ inv norms -----
__global__ __launch_bounds__(256) void uot_norm(
    const float* __restrict__ feat1, const float* __restrict__ feat2,
    float* __restrict__ inv1, float* __restrict__ inv2) {
  const int tid  = threadIdx.x;
  const int lane = tid & 31;
  const int gwave = (blockIdx.x * 256 + tid) >> 5;   // 0..32767
  const float* src; float* dst; int row;
  if (gwave < BATCH * NPTS) { src = feat1; dst = inv1; row = gwave; }
  else                      { src = feat2; dst = inv2; row = gwave - BATCH * NPTS; }
  const float* p = src + (size_t)row * CDIM;
  float s = 0.f;
#pragma unroll
  for (int i = 0; i < CDIM / 32; ++i) { float v = p[lane + 32 * i]; s += v * v; }
#pragma unroll
  for (int off = 16; off > 0; off >>= 1) s += __shfl_down(s, off);
  if (lane == 0) dst[row] = rsqrtf(s + 1e-8f);
}

// ------------------------------------------- K = exp(-(1-cos)/eps)  GEMM ----
// 128x128 block tile, 8 waves (wave tile 32x64 = 2x4 WMMA f32 16x16x4 accs).
// Tiles staged via GLOBAL_LOAD_ASYNC_TO_LDS_B128 with double buffering:
// compute on buffer kc overlaps the async fill of buffer kc+1 (ASYNCcnt).
__global__ __launch_bounds__(256) void uot_gemm_k(
    const float* __restrict__ feat1, const float* __restrict__ feat2,
    const float* __restrict__ inv1, const float* __restrict__ inv2,
    const float* __restrict__ scal, float* __restrict__ Kmat) {
  __shared__ float lds1[2][128][LPAD];
  __shared__ float lds2[2][128][LPAD];
  const int tid  = threadIdx.x;
  const int lane = tid & 31;
  const int wave = tid >> 5;
  const int wr = wave >> 1;        // 0..3 : 32-row sub-tile in N
  const int wc = wave & 1;         // 0..1 : 64-col sub-tile in M
  const int b  = blockIdx.z;
  const int n0 = blockIdx.y * 128;
  const int m0 = blockIdx.x * 128;
  const int h = lane >> 4;         // lane half
  const int r = lane & 15;

  v8f zero = {};
  v8f acc[2][4];
#pragma unroll
  for (int i = 0; i < 2; ++i)
#pragma unroll
    for (int j = 0; j < 4; ++j) acc[i][j] = zero;

  const float* f1 = feat1 + (size_t)(b * NPTS + n0) * CDIM;
  const float* f2 = feat2 + (size_t)(b * MPTS + m0) * CDIM;

  const unsigned l1base = (unsigned)(uintptr_t)&lds1[0][0][0];
  const unsigned l2base = (unsigned)(uintptr_t)&lds2[0][0][0];
  const int row0 = tid >> 2;            // 0..63
  const int kq   = (tid & 3) << 2;      // 0,4,8,12 (float index in row)
  const unsigned BUFB = 128u * LPAD * 4u;   // bytes per buffer

  // issue 4 async B128 loads (2 rows x 2 matrices) for chunk k0 into buf
#define UOT_ISSUE(k0_, buf_)                                                   \
  {                                                                            \
    const unsigned ldso = (unsigned)(buf_) * BUFB + (unsigned)kq * 4u;         \
    const unsigned go   = (unsigned)((k0_) + kq) * 4u;                         \
    _Pragma("unroll")                                                          \
    for (int rr = 0; rr < 2; ++rr) {                                           \
      const unsigned rowb = (unsigned)(row0 + rr * 64);                        \
      const unsigned l1 = l1base + ldso + rowb * (LPAD * 4u);                  \
      const unsigned l2 = l2base + ldso + rowb * (LPAD * 4u);                  \
      const unsigned g  = go + rowb * (CDIM * 4u);                             \
      asm volatile("global_load_async_to_lds_b128 %0, %1, %2"                  \
                   :: "v"(l1), "v"(g), "s"(f1) : "memory");                    \
      asm volatile("global_load_async_to_lds_b128 %0, %1, %2"                  \
                   :: "v"(l2), "v"(g), "s"(f2) : "memory");                    \
    }                                                                          \
  }

  UOT_ISSUE(0, 0)

  for (int kc = 0; kc < CDIM / 16; ++kc) {
    const int buf = kc & 1;
    asm volatile("s_wait_asynccnt 0x0" ::: "memory");  // my chunk-kc loads done
    __syncthreads();                                   // everyone's done; buf^1 free
    if (kc + 1 < CDIM / 16) UOT_ISSUE((kc + 1) * 16, buf ^ 1)

#pragma unroll
    for (int ks = 0; ks < 16; ks += 4) {
      const int kk = ks + 2 * h;                 // lane holds (kk, kk+1)
      v2f afrag[2], bfrag[4];
      afrag[0] = *(const v2f*)&lds1[buf][wr * 32 +      r][kk];
      afrag[1] = *(const v2f*)&lds1[buf][wr * 32 + 16 + r][kk];
#pragma unroll
      for (int mi = 0; mi < 4; ++mi)
        bfrag[mi] = *(const v2f*)&lds2[buf][wc * 64 + mi * 16 + r][kk];
#pragma unroll
      for (int ni = 0; ni < 2; ++ni)
#pragma unroll
        for (int mi = 0; mi < 4; ++mi)
          acc[ni][mi] = __builtin_amdgcn_wmma_f32_16x16x4_f32(
              false, afrag[ni], false, bfrag[mi], 0, acc[ni][mi], false, false);
    }
  }
#undef UOT_ISSUE

  const float inv_eps = 1.0f / scal[0];
  float iv2v[4];
#pragma unroll
  for (int mi = 0; mi < 4; ++mi)
    iv2v[mi] = inv2[b * MPTS + m0 + wc * 64 + mi * 16 + r];
#pragma unroll
  for (int ni = 0; ni < 2; ++ni) {
#pragma unroll
    for (int v = 0; v < 8; ++v) {
      int n = n0 + wr * 32 + ni * 16 + 8 * h + v;   // D row = vgpr + 8*half
      float iv1v = inv1[b * NPTS + n];
#pragma unroll
      for (int mi = 0; mi < 4; ++mi) {
        int m = m0 + wc * 64 + mi * 16 + r;         // D col = lane&15
        float dot  = acc[ni][mi][v];
        float cost = 1.0f - dot * iv1v * iv2v[mi];
        Kmat[(size_t)(b * NPTS + n) * MPTS + m] = __expf(-cost * inv_eps);
      }
    }
  }
}

// ---------------------------------------------- KTa = K^T a ; b update ------
__global__ __launch_bounds__(256) void uot_kta_b(
    const float* __restrict__ Kmat, const float* __restrict__ avec,
    const float* __restrict__ scal, float* __restrict__ bvec) {
  __shared__ float sa[NPTS];
  const int tid = threadIdx.x;
  const int b = blockIdx.x >> 3;
  const int m = ((blockIdx.x & 7) << 8) + tid;
  for (int i = tid; i < NPTS; i += 256) sa[i] = avec[(b << 11) + i];
  __syncthreads();
  const float* Kp = Kmat + (size_t)b * NPTS * MPTS + m;
  float s0 = 0.f, s1 = 0.f, s2 = 0.f, s3 = 0.f;
  for (int n = 0; n < NPTS; n += 4) {
    s0 += Kp[(size_t)(n + 0) * MPTS] * sa[n + 0];
    s1 += Kp[(size_t)(n + 1) * MPTS] * sa[n + 1];
    s2 += Kp[(size_t)(n + 2) * MPTS] * sa[n + 2];
    s3 += Kp[(size_t)(n + 3) * MPTS] * sa[n + 3];
  }
  float kta = (s0 + s1) + (s2 + s3);
  bvec[(b << 11) + m] = powf((1.0f / MPTS) / (kta + 1e-8f), scal[1]);
}

// ---------------------------------------------- Kb = K b ; a update ---------
__global__ __launch_bounds__(256) void uot_kb_a(
    const float* __restrict__ Kmat, const float* __restrict__ bvec,
    const float* __restrict__ scal, float* __restrict__ avec,
    float* __restrict__ Kb) {
  __shared__ float sb[MPTS];
  const int tid  = threadIdx.x;
  const int lane = tid & 31;
  const int wave = tid >> 5;
  const int b = blockIdx.x >> 8;
  const int n = ((blockIdx.x & 255) << 3) + wave;
  for (int i = tid; i < MPTS; i += 256) sb[i] = bvec[(b << 11) + i];
  __syncthreads();
  const float* Kr = Kmat + (size_t)(b * NPTS + n) * MPTS;
  float s = 0.f;
  for (int m = lane; m < MPTS; m += 32) s += Kr[m] * sb[m];
#pragma unroll
  for (int off = 16; off > 0; off >>= 1) s += __shfl_down(s, off);
  if (lane == 0) {
    int row = (b << 11) + n;
    Kb[row]   = s;
    avec[row] = powf((1.0f / NPTS) / (s + 1e-8f), scal[1]);
  }
}

// ----------------------- row_sum + soft correspondences (T @ coords2) -------
__global__ __launch_bounds__(256) void uot_corr(
    const float* __restrict__ Kmat, const float* __restrict__ bvec,
    const float* __restrict__ coords2, const float* __restrict__ avec,
    const float* __restrict__ Kb, float* __restrict__ corr,
    float* __restrict__ rs) {
  __shared__ float sb[MPTS];
  __shared__ float sc2[MPTS * 3];
  const int tid  = threadIdx.x;
  const int lane = tid & 31;
  const int wave = tid >> 5;
  const int b = blockIdx.x >> 8;
  const int n = ((blockIdx.x & 255) << 3) + wave;
  for (int i = tid; i < MPTS; i += 256) sb[i] = bvec[(b << 11) + i];
  for (int i = tid; i < MPTS * 3; i += 256) sc2[i] = coords2[(size_t)b * MPTS * 3 + i];
  __syncthreads();
  const float* Kr = Kmat + (size_t)(b * NPTS + n) * MPTS;
  float sx = 0.f, sy = 0.f, sz = 0.f;
  for (int m = lane; m < MPTS; m += 32) {
    float kb = Kr[m] * sb[m];
    sx += kb * sc2[m * 3 + 0];
    sy += kb * sc2[m * 3 + 1];
    sz += kb * sc2[m * 3 + 2];
  }
#pragma unroll
  for (int off = 16; off > 0; off >>= 1) {
    sx += __shfl_down(sx, off);
    sy += __shfl_down(sy, off);
    sz += __shfl_down(sz, off);
  }
  if (lane == 0) {
    int row = (b << 11) + n;
    float af = avec[row], kbv = Kb[row];
    float rsum = af * kbv;              // a[n] * (K b)[n]
    rs[row] = rsum;
    float inv = af / (rsum + 1e-8f);
    corr[row * 3 + 0] = sx * inv;
    corr[row * 3 + 1] = sy * inv;
    corr[row * 3 + 2] = sz * inv;
  }
}

// --------------------------------- per-batch centroids + 3x3 covariance ----
__device__ __forceinline__ float uot_block_reduce(float v, float* red, int tid) {
  red[tid] = v; __syncthreads();
  for (int s = 128; s > 0; s >>= 1) { if (tid < s) red[tid] += red[tid + s]; __syncthreads(); }
  float r = red[0]; __syncthreads();
  return r;
}

__global__ __launch_bounds__(256) void uot_stats(
    const float* __restrict__ coords1, const float* __restrict__ corr,
    const float* __restrict__ rs, float* __restrict__ stats) {
  __shared__ float red[256];
  const int tid = threadIdx.x;
  const int b = blockIdx.x;
  float sw = 0.f, s1[3] = {0, 0, 0}, s2[3] = {0, 0, 0};
  for (int n = tid; n < NPTS; n += 256) {
    int row = (b << 11) + n;
    float w = rs[row];
    sw += w;
    const float* c1 = coords1 + (size_t)row * 3;
    const float* cr = corr + (size_t)row * 3;
#pragma unroll
    for (int c = 0; c < 3; ++c) { s1[c] += w * c1[c]; s2[c] += w * cr[c]; }
  }
  sw = uot_block_reduce(sw, red, tid);
  float invw = 1.0f / (sw + 1e-5f);
  float ca[3], cb[3];
#pragma unroll
  for (int c = 0; c < 3; ++c) ca[c] = uot_block_reduce(s1[c], red, tid) * invw;
#pragma unroll
  for (int c = 0; c < 3; ++c) cb[c] = uot_block_reduce(s2[c], red, tid) * invw;

  float cov[9] = {0, 0, 0, 0, 0, 0, 0, 0, 0};
  for (int n = tid; n < NPTS; n += 256) {
    int row = (b << 11) + n;
    float wn = rs[row] * invw;
    const float* c1 = coords1 + (size_t)row * 3;
    const float* cr = corr + (size_t)row * 3;
    float dx[3], dy[3];
#pragma unroll
    for (int c = 0; c < 3; ++c) { dx[c] = c1[c] - ca[c]; dy[c] = (cr[c] - cb[c]) * wn; }
#pragma unroll
    for (int i = 0; i < 3; ++i)
#pragma unroll
      for (int j = 0; j < 3; ++j) cov[i * 3 + j] += dx[i] * dy[j];
  }
#pragma unroll
  for (int k = 0; k < 9; ++k) cov[k] = uot_block_reduce(cov[k], red, tid);
  if (tid == 0) {
    float* st = stats + b * 16;
#pragma unroll
    for (int k = 0; k < 9; ++k) st[k] = cov[k];
#pragma unroll
    for (int c = 0; c < 3; ++c) { st[9 + c] = ca[c]; st[12 + c] = cb[c]; }
  }
}

// ----------------------------------- 3x3 SVD (Jacobi) + rigid transform ----
__device__ __forceinline__ void uot_jrot(float S[3][3], float V[3][3], int p, int q) {
  float apq = S[p][q];
  if (fabsf(apq) < 1e-20f) return;
  float tau = (S[q][q] - S[p][p]) / (2.0f * apq);
  float t   = copysignf(1.0f, tau) / (fabsf(tau) + sqrtf(1.0f + tau * tau));
  float c   = rsqrtf(1.0f + t * t);
  float sn  = t * c;
#pragma unroll
  for (int k = 0; k < 3; ++k) {
    float skp = S[k][p], skq = S[k][q];
    S[k][p] = c * skp - sn * skq;
    S[k][q] = sn * skp + c * skq;
  }
#pragma unroll
  for (int k = 0; k < 3; ++k) {
    float spk = S[p][k], sqk = S[q][k];
    S[p][k] = c * spk - sn * sqk;
    S[q][k] = sn * spk + c * sqk;
  }
#pragma unroll
  for (int k = 0; k < 3; ++k) {
    float vkp = V[k][p], vkq = V[k][q];
    V[k][p] = c * vkp - sn * vkq;
    V[k][q] = sn * vkp + c * vkq;
  }
}

__global__ void uot_svd(const float* __restrict__ stats, float* __restrict__ out) {
  int b = threadIdx.x;
  if (b >= BATCH) return;
  const float* st = stats + b * 16;
  float A[3][3], ca[3], cb[3];
#pragma unroll
  for (int i = 0; i < 3; ++i)
#pragma unroll
    for (int j = 0; j < 3; ++j) A[i][j] = st[i * 3 + j];
#pragma unroll
  for (int c = 0; c < 3; ++c) { ca[c] = st[9 + c]; cb[c] = st[12 + c]; }

  float S[3][3], V[3][3] = {{1, 0, 0}, {0, 1, 0}, {0, 0, 1}};
#pragma unroll
  for (int i = 0; i < 3; ++i)
#pragma unroll
    for (int j = 0; j < 3; ++j)
      S[i][j] = A[0][i] * A[0][j] + A[1][i] * A[1][j] + A[2][i] * A[2][j];

  for (int sweep = 0; sweep < 15; ++sweep) {
    uot_jrot(S, V, 0, 1);
    uot_jrot(S, V, 0, 2);
    uot_jrot(S, V, 1, 2);
  }
  float lam[3] = {S[0][0], S[1][1], S[2][2]};
#pragma unroll
  for (int i = 0; i < 2; ++i)
#pragma unroll
    for (int j = 0; j < 2 - i; ++j)
      if (lam[j] < lam[j + 1]) {
        float tl = lam[j]; lam[j] = lam[j + 1]; lam[j + 1] = tl;
#pragma unroll
        for (int k = 0; k < 3; ++k) { float tv = V[k][j]; V[k][j] = V[k][j + 1]; V[k][j + 1] = tv; }
      }
  float U[3][3];
#pragma unroll
  for (int i = 0; i < 3; ++i) {
    float sig  = sqrtf(fmaxf(lam[i], 0.0f));
    float invs = 1.0f / fmaxf(sig, 1e-12f);
#pragma unroll
    for (int rr = 0; rr < 3; ++rr)
      U[rr][i] = (A[rr][0] * V[0][i] + A[rr][1] * V[1][i] + A[rr][2] * V[2][i]) * invs;
  }
  float R[3][3];
#pragma unroll
  for (int i = 0; i < 3; ++i)
#pragma unroll
    for (int j = 0; j < 3; ++j)
      R[i][j] = V[i][0] * U[j][0] + V[i][1] * U[j][1] + V[i][2] * U[j][2];
  float det = R[0][0] * (R[1][1] * R[2][2] - R[1][2] * R[2][1])
            - R[0][1] * (R[1][0] * R[2][2] - R[1][2] * R[2][0])
            + R[0][2] * (R[1][0] * R[2][1] - R[1][1] * R[2][0]);
  if (!(det > 0.0f)) {
#pragma unroll
    for (int i = 0; i < 3; ++i)
#pragma unroll
      for (int j = 0; j < 3; ++j) R[i][j] -= 2.0f * V[i][2] * U[j][2];
  }
  float tv[3];
#pragma unroll
  for (int rr = 0; rr < 3; ++rr)
    tv[rr] = -(R[rr][0] * ca[0] + R[rr][1] * ca[1] + R[rr][2] * ca[2]) + cb[rr];
#pragma unroll
  for (int rr = 0; rr < 3; ++rr) {
#pragma unroll
    for (int cc = 0; cc < 3; ++cc) out[b * 12 + rr * 4 + cc] = R[rr][cc];
    out[b * 12 + rr * 4 + 3] = tv[rr];
  }
}

// ---------------------------------------------------------------------------
extern "C" void kernel_launch(void* const* d_in, const int* in_sizes, int n_in,
                              void* d_out, int out_size, void* d_ws, size_t ws_size,
                              hipStream_t stream) {
  (void)in_sizes; (void)n_in; (void)out_size; (void)ws_size;
  const float* feat1     = (const float*)d_in[0];
  const float* feat2     = (const float*)d_in[1];
  const float* coords1   = (const float*)d_in[2];
  const float* coords2   = (const float*)d_in[3];
  const float* log_eps   = (const float*)d_in[4];
  const float* log_gamma = (const float*)d_in[5];
  float* out = (float*)d_out;

  float* W = (float*)d_ws;
  const size_t KSZ = (size_t)BATCH * NPTS * MPTS;   // 128 MB — L2-resident
  float* Kmat  = W;
  float* inv1  = Kmat + KSZ;
  float* inv2  = inv1 + BATCH * NPTS;
  float* avec  = inv2 + BATCH * MPTS;
  float* bvec  = avec + BATCH * NPTS;
  float* Kb    = bvec + BATCH * MPTS;
  float* rs    = Kb + BATCH * NPTS;
  float* corr  = rs + BATCH * NPTS;
  float* stats = corr + (size_t)BATCH * NPTS * 3;
  float* scal  = stats + BATCH * 16;

  uot_init<<<64, 256, 0, stream>>>(log_eps, log_gamma, avec, scal);
  uot_norm<<<4096, 256, 0, stream>>>(feat1, feat2, inv1, inv2);
  uot_gemm_k<<<dim3(16, 16, 8), 256, 0, stream>>>(feat1, feat2, inv1, inv2, scal, Kmat);
  for (int it = 0; it < UOT_ITERS; ++it) {
    uot_kta_b<<<64, 256, 0, stream>>>(Kmat, avec, scal, bvec);
    uot_kb_a<<<2048, 256, 0, stream>>>(Kmat, bvec, scal, avec, Kb);
  }
  uot_corr<<<2048, 256, 0, stream>>>(Kmat, bvec, coords2, avec, Kb, corr, rs);
  uot_stats<<<8, 256, 0, stream>>>(coords1, corr, rs, stats);
  uot_svd<<<1, 32, 0, stream>>>(stats, out);
}